// TF2FullModel_65730179498165
// MI455X (gfx1250) — compile-verified
//
#include <hip/hip_runtime.h>
#include <math.h>

typedef __attribute__((ext_vector_type(2))) float v2f;
typedef __attribute__((ext_vector_type(8))) float v8f;

#define NNODES 50000
#define NEDGES 800000
#define FDIM   128
#define UDIM   128
#define FCDIM  256
#define CDIM   40

// ---------------------------------------------------------------------------
// Dense GEMM on the CDNA5 matrix pipe: Y[:, colOff:colOff+cols] (leading dim
// ldY) = relu(X[N,K] @ W[K,cols] + bias).  fp32 end-to-end via
// V_WMMA_F32_16X16X4_F32.  128 threads = 4 waves; block stages a 16xK tile of
// X into LDS, each wave owns one 16-wide column tile.
// Requires: N % 16 == 0, K % 4 == 0, cols % 64 == 0 (all true here).
// ---------------------------------------------------------------------------
__global__ void gemm_wmma_f32(const float* __restrict__ X,
                              const float* __restrict__ W,
                              const float* __restrict__ bias,
                              float* __restrict__ Y,
                              int K, int cols, int ldY, int colOff, int doRelu)
{
  extern __shared__ float xs[];           // [16][K+4] padded to dodge bank conflicts
  const int ldx  = K + 4;
  const int row0 = blockIdx.x * 16;
  const int tid  = threadIdx.x;

  // Cooperative load of the 16 x K activation tile.
  for (int idx = tid; idx < 16 * K; idx += 128) {
    int r = idx / K;
    int c = idx - r * K;
    xs[r * ldx + c] = X[(size_t)(row0 + r) * K + c];
  }
  __syncthreads();

  const int wave = tid >> 5;
  const int lane = tid & 31;
  const int half = lane >> 4;       // 0: K pair {0,1} / M 0..7 ; 1: {2,3} / M 8..15
  const int mn   = lane & 15;       // A-row / B-col / C-col within tile
  const int ksel = half << 1;
  const int col  = blockIdx.y * 64 + wave * 16 + mn;

  // Fold bias into the initial accumulator (bias depends only on the column).
  const float bv = bias[col];
  v8f acc;
#pragma unroll
  for (int r = 0; r < 8; ++r) acc[r] = bv;

  const float* xrow = xs + mn * ldx;
#pragma unroll 4
  for (int kk = 0; kk < K; kk += 4) {
    v2f a, b;
    a.x = xrow[kk + ksel];                       // A 16x4 fragment (LDS)
    a.y = xrow[kk + ksel + 1];
    b.x = W[(size_t)(kk + ksel) * cols + col];   // B 4x16 fragment (L2-resident W)
    b.y = W[(size_t)(kk + ksel + 1) * cols + col];
    acc = __builtin_amdgcn_wmma_f32_16x16x4_f32(
        /*neg_a=*/false, a, /*neg_b=*/false, b,
        /*c_mod=*/(short)0, acc, /*reuse_a=*/false, /*reuse_b=*/false);
  }

  const int mbase = half * 8;                    // C layout: VGPR r -> M = r (+8 for upper lanes)
#pragma unroll
  for (int r = 0; r < 8; ++r) {
    float v = acc[r];
    if (doRelu) v = fmaxf(v, 0.0f);
    Y[(size_t)(row0 + mbase + r) * ldY + colOff + col] = v;
  }
}

// ---------------------------------------------------------------------------
// SPMM  Y = A @ X  (edges src->dst, weight w).  One wave per edge; lanes
// stride the feature dim (coalesced gather from L2-resident X, float atomic
// scatter-add).  Y must be pre-zeroed.
// ---------------------------------------------------------------------------
__global__ void spmm_atomic(const int* __restrict__ src,
                            const int* __restrict__ dst,
                            const float* __restrict__ w,
                            const float* __restrict__ X,
                            float* __restrict__ Y,
                            int nedges, int D)
{
  const int lane   = threadIdx.x & 31;
  const int gwave  = (blockIdx.x * blockDim.x + threadIdx.x) >> 5;
  const int nwaves = (gridDim.x * blockDim.x) >> 5;
  for (int e = gwave; e < nedges; e += nwaves) {
    const int   s  = src[e];
    const int   d  = dst[e];
    const float wt = w[e];
    const float* xp = X + (size_t)s * D;
    float*       yp = Y + (size_t)d * D;
    for (int f = lane; f < D; f += 32)
      atomicAdd(&yp[f], wt * xp[f]);
  }
}

// ---------------------------------------------------------------------------
// Fused output head: logits = y @ W_out + b_out, then row softmax.
// One wave32 per node; lane covers classes {lane, lane+32}.
// ---------------------------------------------------------------------------
__global__ void out_softmax(const float* __restrict__ Yfc,
                            const float* __restrict__ Wout,
                            const float* __restrict__ bout,
                            float* __restrict__ P, int n)
{
  const int gwave = (blockIdx.x * blockDim.x + threadIdx.x) >> 5;
  const int lane  = threadIdx.x & 31;
  if (gwave >= n) return;

  const float* yrow = Yfc + (size_t)gwave * FCDIM;
  const int  c0   = lane;
  const int  c1   = lane + 32;
  const bool has1 = (c1 < CDIM);

  float a0 = bout[c0];
  float a1 = has1 ? bout[c1] : 0.0f;
  for (int k = 0; k < FCDIM; ++k) {
    const float xv = yrow[k];                    // uniform across lanes
    a0 = fmaf(xv, Wout[(size_t)k * CDIM + c0], a0);
    if (has1) a1 = fmaf(xv, Wout[(size_t)k * CDIM + c1], a1);
  }

  float m = has1 ? fmaxf(a0, a1) : a0;
  for (int o = 16; o > 0; o >>= 1) m = fmaxf(m, __shfl_xor(m, o, 32));
  const float e0 = __expf(a0 - m);
  const float e1 = has1 ? __expf(a1 - m) : 0.0f;
  float s = e0 + e1;
  for (int o = 16; o > 0; o >>= 1) s += __shfl_xor(s, o, 32);
  const float inv = 1.0f / s;

  P[(size_t)gwave * CDIM + c0] = e0 * inv;
  if (has1) P[(size_t)gwave * CDIM + c1] = e1 * inv;
}

// ---------------------------------------------------------------------------
extern "C" void kernel_launch(void* const* d_in, const int* in_sizes, int n_in,
                              void* d_out, int out_size, void* d_ws, size_t ws_size,
                              hipStream_t stream)
{
  const int*   esrc = (const int*)d_in[0];
  const int*   edst = (const int*)d_in[1];
  const float* ew   = (const float*)d_in[2];
  const float* H    = (const float*)d_in[3];
  const float* Wb[2][3] = {
    { (const float*)d_in[4],  (const float*)d_in[6],  (const float*)d_in[8]  },
    { (const float*)d_in[10], (const float*)d_in[12], (const float*)d_in[14] } };
  const float* bb[2][3] = {
    { (const float*)d_in[5],  (const float*)d_in[7],  (const float*)d_in[9]  },
    { (const float*)d_in[11], (const float*)d_in[13], (const float*)d_in[15] } };
  const float* Wfc  = (const float*)d_in[16];
  const float* bfc  = (const float*)d_in[17];
  const float* Wout = (const float*)d_in[18];
  const float* bout = (const float*)d_in[19];

  // Workspace: cur0 [N,384] | P1 [N,384] | P2 [N,384]; yfc aliases P1 (dead by then).
  float* cur0 = (float*)d_ws;
  float* P1   = cur0 + (size_t)NNODES * 384;
  float* P2   = P1   + (size_t)NNODES * 384;
  float* yfc  = P1;

  float* out_cur = (float*)d_out;                   // [N, 384]
  float* out_p   = out_cur + (size_t)NNODES * 384;  // [N, 40]

  const dim3 blk(128);
  auto gemm = [&](const float* X, int K, const float* W, const float* b,
                  int cols, float* Y, int ldY, int colOff) {
    dim3 grid(NNODES / 16, cols / 64);
    size_t smem = (size_t)16 * (K + 4) * sizeof(float);
    gemm_wmma_f32<<<grid, blk, smem, stream>>>(X, W, b, Y, K, cols, ldY, colOff, 1);
  };
  auto spmm = [&](const float* X, int D, float* Y) {
    hipMemsetAsync(Y, 0, (size_t)NNODES * D * sizeof(float), stream);
    spmm_atomic<<<4096, 256, 0, stream>>>(esrc, edst, ew, X, Y, NEDGES, D);
  };

  // ---- Block 0 (input F=128) -> cur0 [N, 3*128]
  gemm(H,  FDIM, Wb[0][0], bb[0][0], UDIM, cur0, 384, 0);
  spmm(H,  FDIM, P1);
  gemm(P1, FDIM, Wb[0][1], bb[0][1], UDIM, cur0, 384, 128);
  spmm(P1, FDIM, P2);
  gemm(P2, FDIM, Wb[0][2], bb[0][2], UDIM, cur0, 384, 256);

  // ---- Block 1 (input 384) -> out_cur [N, 3*128]  (first output of the tuple)
  gemm(cur0, 384, Wb[1][0], bb[1][0], UDIM, out_cur, 384, 0);
  spmm(cur0, 384, P1);
  gemm(P1,   384, Wb[1][1], bb[1][1], UDIM, out_cur, 384, 128);
  spmm(P1,   384, P2);
  gemm(P2,   384, Wb[1][2], bb[1][2], UDIM, out_cur, 384, 256);

  // ---- FC + softmax head -> out_p [N, 40]
  gemm(out_cur, 384, Wfc, bfc, FCDIM, yfc, FCDIM, 0);
  const int waves_per_block = 256 / 32;
  const int sblocks = (NNODES + waves_per_block - 1) / waves_per_block;
  out_softmax<<<sblocks, 256, 0, stream>>>(yfc, Wout, bout, out_p, NNODES);
}